// UOTPooling_65386582114529
// MI455X (gfx1250) — compile-verified
//
#include <hip/hip_runtime.h>
#include <hip/hip_bf16.h>
#include <math.h>
#include <stdint.h>

typedef __attribute__((ext_vector_type(2))) float v2f;
typedef __attribute__((ext_vector_type(8))) float v8f;
typedef __attribute__((ext_vector_type(4))) unsigned int u32x4;
typedef __attribute__((ext_vector_type(8))) int i32x8;
typedef __attribute__((ext_vector_type(4))) int i32x4;

#define DF 128   // feature dim (d)
#define NT 8     // DF / 16 feature tiles
#define TN 16    // nodes per TDM tile (16 rows x 128 f32 = 8KB)

__device__ __forceinline__ float softplus_f(float v) {
    return fmaxf(v, 0.0f) + log1pf(__expf(-fabsf(v)));
}

// Issue one TDM load: rows_tile x 128 f32 tile at gsrc into LDS at byte offset lds_off.
// D# per CDNA5 ISA ch.8: group0 = count/lds/global/type; group1 = dims/strides.
__device__ __forceinline__ void tdm_load_tile(const float* gsrc, unsigned lds_off,
                                              int rows_avail, int rows_tile) {
    unsigned long long ga = (unsigned long long)(uintptr_t)gsrc;
    u32x4 g0;
    g0.x = 1u;                                           // count=1, user mode
    g0.y = lds_off;                                      // lds_addr (bytes)
    g0.z = (unsigned)(ga & 0xFFFFFFFFu);                 // global_addr[31:0]
    g0.w = (unsigned)((ga >> 32) & 0x01FFFFFFu) | (2u << 30); // addr[56:32]|type=2
    i32x8 g1;
    g1[0] = 0x00020000;                                  // data_size=4B, no flags
    g1[1] = (int)(((unsigned)DF & 0xFFFFu) << 16);       // tensor_dim0 lo16
    g1[2] = (int)((((unsigned)rows_avail) & 0xFFFFu) << 16); // dim0 hi=0 | dim1 lo16
    g1[3] = (int)((((unsigned)rows_avail >> 16) & 0xFFFFu)
                  | (((unsigned)DF & 0xFFFFu) << 16));   // dim1 hi | tile_dim0=128
    g1[4] = (int)((unsigned)rows_tile & 0xFFFFu);        // tile_dim1 | tile_dim2=0
    g1[5] = DF;                                          // tensor_dim0_stride lo32
    g1[6] = 0;                                           // stride hi | dim1_stride lo
    g1[7] = 0;
    i32x4 gz4 = {0, 0, 0, 0};                            // 2D tile: groups 2/3 unused
    i32x8 gz8 = {0, 0, 0, 0, 0, 0, 0, 0};
    __builtin_amdgcn_tensor_load_to_lds(g0, g1, gz4, gz4, gz8, 0);
}

// ---------------------------------------------------------------- utilities
__global__ void zero_kernel(unsigned int* p, long n) {
    long i = (long)blockIdx.x * blockDim.x + threadIdx.x;
    if (i < n) p[i] = 0u;
}

__global__ void count_kernel(const int* __restrict__ batch,
                             int* __restrict__ counts, int n) {
    int i = blockIdx.x * blockDim.x + threadIdx.x;
    if (i < n) atomicAdd(&counts[batch[i]], 1);
}

// exclusive scan of counts -> starts (B <= 1024), single block
__global__ void scan_kernel(const int* __restrict__ counts,
                            int* __restrict__ starts, int B) {
    __shared__ int tmp[1024];
    int t = threadIdx.x;
    tmp[t] = (t < B) ? counts[t] : 0;
    __syncthreads();
    for (int off = 1; off < 1024; off <<= 1) {
        int v = (t >= off) ? tmp[t - off] : 0;
        __syncthreads();
        tmp[t] += v;
        __syncthreads();
    }
    if (t < B) starts[t + 1] = tmp[t];
    if (t == 0) starts[0] = 0;
}

// coef layout: [0..K) 1/a1[k-1]; [K..2K) c1; [2K..3K) c2; [3K..4K) c3;
//              [4K..5K) c4; [5K] = 1/a1[K-1]
__global__ void coef_kernel(const float* __restrict__ a1p,
                            const float* __restrict__ a2p,
                            const float* __restrict__ a3p,
                            float* __restrict__ coef, int K) {
    int k = blockIdx.x * blockDim.x + threadIdx.x;
    if (k >= K) return;
    float a1k = softplus_f(a1p[k]);
    float a2k = softplus_f(a2p[k]);
    float a3k = softplus_f(a3p[k]);
    float a1prev = softplus_f(a1p[k > 0 ? k - 1 : 0]);
    coef[k]         = 1.0f / a1prev;
    coef[K + k]     = a2k / (a1k * (a1k + a2k));
    coef[2 * K + k] = a2k / (a1k + a2k);
    coef[3 * K + k] = a3k / (a1k * (a1k + a3k));
    coef[4 * K + k] = a3k / (a1k + a3k);
    if (k == K - 1) coef[5 * K] = 1.0f / a1k;
}

// ---------------------------------------------------------------- one scan step
// One block (4 waves) per segment. TDM double-buffers 16-node x-tiles into LDS;
// column sums over nodes via f32 WMMA (A = ones 16x4 -> D rows = column sums);
// per-node feature sums via 16-lane shuffle trees.
__global__ __launch_bounds__(128) void iter_kernel(
    const float* __restrict__ x, const int* __restrict__ counts,
    const int* __restrict__ starts, float* __restrict__ b1,
    float* __restrict__ b2, const float* __restrict__ coef, int K, int k)
{
    const int seg  = blockIdx.x;
    const int lane = threadIdx.x & 31;
    const int wv   = threadIdx.x >> 5;
    const int fcol = lane & 15;
    const int hi   = lane >> 4;

    const int start = starts[seg];
    const int end   = starts[seg + 1];
    const float cnt = (float)counts[seg];

    const float inv_a1p = coef[k];
    const float c1 = coef[K + k];
    const float c2 = coef[2 * K + k];
    const float c3 = coef[3 * K + k];
    const float c4 = coef[4 * K + k];
    const float logu1 = -__logf(cnt);            // -log(count)  (unused if empty)
    const float logu2 = -__logf((float)DF);      // -log(d)

    __shared__ float smem[2 * TN * DF + 4 * DF]; // 2 tile buffers + wave partials
    float* sS = smem + 2 * TN * DF;
    const unsigned lds_base = (unsigned)(uintptr_t)(void*)smem; // low 32 bits are the LDS offset

    float b2l[NT];
    #pragma unroll
    for (int ft = 0; ft < NT; ++ft) b2l[ft] = b2[seg * DF + ft * 16 + fcol];

    v8f acc[NT];
    #pragma unroll
    for (int ft = 0; ft < NT; ++ft) {
        v8f zv = {0.f, 0.f, 0.f, 0.f, 0.f, 0.f, 0.f, 0.f};
        acc[ft] = zv;
    }
    const v2f ones = {1.0f, 1.0f};

    if (wv == 0 && start < end) {
        int rem = end - start;
        tdm_load_tile(x + (size_t)start * DF, lds_base, rem, rem < TN ? rem : TN);
    }

    int bi = 0;
    for (int g = start; g < end; g += TN) {           // block-uniform loop
        if (wv == 0) __builtin_amdgcn_s_wait_tensorcnt(0);  // tile g resident
        __syncthreads();
        int gn = g + TN;
        if (wv == 0 && gn < end) {                    // prefetch next tile via TDM
            int rem = end - gn;
            tdm_load_tile(x + (size_t)gn * DF,
                          lds_base + (unsigned)((bi ^ 1) * TN * DF * 4),
                          rem, rem < TN ? rem : TN);
        }
        const float* buf = smem + bi * TN * DF;

        int r0 = wv * 4;                 // this wave's 4 rows of the 16-row tile
        int nA = g + r0 + hi;            // nodes r0, r0+1 across wave halves
        int nB = g + r0 + 2 + hi;        // nodes r0+2, r0+3
        bool vA = nA < end, vB = nB < end;
        int nAc = vA ? nA : end - 1;
        int nBc = vB ? nB : end - 1;
        float b1A = b1[nAc];
        float b1B = b1[nBc];
        float rsA = 0.0f, rsB = 0.0f;
        #pragma unroll
        for (int ft = 0; ft < NT; ++ft) {
            float xA = buf[(r0 + hi) * DF + ft * 16 + fcol];
            float xB = buf[(r0 + 2 + hi) * DF + ft * 16 + fcol];
            float tA = xA * inv_a1p;
            float tB = xB * inv_a1p;
            // column-sum argument: exp(x/a1_prev + b1_prev[i])
            float eA = __expf(tA + b1A);
            float eB = __expf(tB + b1B);
            v2f bb;
            bb.x = vA ? eA : 0.0f;
            bb.y = vB ? eB : 0.0f;
            acc[ft] = __builtin_amdgcn_wmma_f32_16x16x4_f32(
                false, ones, false, bb, (short)0, acc[ft], false, false);
            // row-sum argument: exp(x/a1_prev + b2_prev[s,f])
            float rA = __expf(tA + b2l[ft]);
            float rB = __expf(tB + b2l[ft]);
            rsA += vA ? rA : 0.0f;
            rsB += vB ? rB : 0.0f;
        }
        #pragma unroll
        for (int off = 1; off < 16; off <<= 1) {
            rsA += __shfl_xor(rsA, off, 32);
            rsB += __shfl_xor(rsB, off, 32);
        }
        if (fcol == 0) {  // lanes 0 and 16 hold their node's full feature sum
            if (vA) { float lm1 = b1A + __logf(rsA);
                      b1[nA] = c1 * b1A + c2 * (logu1 - lm1); }
            if (vB) { float lm1 = b1B + __logf(rsB);
                      b1[nB] = c1 * b1B + c2 * (logu1 - lm1); }
        }
        bi ^= 1;
    }

    // combine wave-level column sums, update b2 (this block owns the row)
    if (lane < 16) {
        #pragma unroll
        for (int ft = 0; ft < NT; ++ft)
            sS[wv * DF + ft * 16 + lane] = acc[ft][0];  // D row 0 = col sum
    }
    __syncthreads();
    int f = threadIdx.x;  // 0..127
    float S = sS[f] + sS[DF + f] + sS[2 * DF + f] + sS[3 * DF + f];
    float b2f = b2[seg * DF + f];
    float lm2 = (cnt > 0.0f) ? (b2f + __logf(S > 0.0f ? S : 1.0f)) : 0.0f;
    b2[seg * DF + f] = c3 * b2f + c4 * (logu2 - lm2);
}

// ---------------------------------------------------------------- epilogue
__global__ __launch_bounds__(128) void final_kernel(
    const float* __restrict__ x, const int* __restrict__ starts,
    const float* __restrict__ b1, const float* __restrict__ b2,
    const float* __restrict__ coef, int K,
    float* __restrict__ z, float* __restrict__ trans)
{
    const int seg  = blockIdx.x;
    const int lane = threadIdx.x & 31;
    const int wv   = threadIdx.x >> 5;
    const int fcol = lane & 15;
    const int hi   = lane >> 4;
    const int start = starts[seg];
    const int end   = starts[seg + 1];
    const float inv_a1 = coef[5 * K];

    __shared__ float smem[2 * TN * DF + 4 * DF];
    float* sS = smem + 2 * TN * DF;
    const unsigned lds_base = (unsigned)(uintptr_t)(void*)smem;

    float b2l[NT];
    #pragma unroll
    for (int ft = 0; ft < NT; ++ft) b2l[ft] = b2[seg * DF + ft * 16 + fcol];

    v8f acc[NT];
    #pragma unroll
    for (int ft = 0; ft < NT; ++ft) {
        v8f zv = {0.f, 0.f, 0.f, 0.f, 0.f, 0.f, 0.f, 0.f};
        acc[ft] = zv;
    }
    const v2f ones = {1.0f, 1.0f};

    if (wv == 0 && start < end) {
        int rem = end - start;
        tdm_load_tile(x + (size_t)start * DF, lds_base, rem, rem < TN ? rem : TN);
    }

    int bi = 0;
    for (int g = start; g < end; g += TN) {
        if (wv == 0) __builtin_amdgcn_s_wait_tensorcnt(0);
        __syncthreads();
        int gn = g + TN;
        if (wv == 0 && gn < end) {
            int rem = end - gn;
            tdm_load_tile(x + (size_t)gn * DF,
                          lds_base + (unsigned)((bi ^ 1) * TN * DF * 4),
                          rem, rem < TN ? rem : TN);
        }
        const float* buf = smem + bi * TN * DF;

        int r0 = wv * 4;
        int nA = g + r0 + hi;
        int nB = g + r0 + 2 + hi;
        bool vA = nA < end, vB = nB < end;
        int nAc = vA ? nA : end - 1;
        int nBc = vB ? nB : end - 1;
        float b1A = b1[nAc], b1B = b1[nBc];
        #pragma unroll
        for (int ft = 0; ft < NT; ++ft) {
            float xA = buf[(r0 + hi) * DF + ft * 16 + fcol];
            float xB = buf[(r0 + 2 + hi) * DF + ft * 16 + fcol];
            float trA = __expf(xA * inv_a1 + b1A + b2l[ft]) + 1e-8f;
            float trB = __expf(xB * inv_a1 + b1B + b2l[ft]) + 1e-8f;
            if (vA) trans[(size_t)nA * DF + ft * 16 + fcol] = trA;
            if (vB) trans[(size_t)nB * DF + ft * 16 + fcol] = trB;
            v2f bb;
            bb.x = vA ? xA * trA : 0.0f;
            bb.y = vB ? xB * trB : 0.0f;
            acc[ft] = __builtin_amdgcn_wmma_f32_16x16x4_f32(
                false, ones, false, bb, (short)0, acc[ft], false, false);
        }
        bi ^= 1;
    }

    if (lane < 16) {
        #pragma unroll
        for (int ft = 0; ft < NT; ++ft)
            sS[wv * DF + ft * 16 + lane] = acc[ft][0];
    }
    __syncthreads();
    int f = threadIdx.x;
    float S = sS[f] + sS[DF + f] + sS[2 * DF + f] + sS[3 * DF + f];
    z[seg * DF + f] = (float)DF * S;
}

// ---------------------------------------------------------------- launcher
extern "C" void kernel_launch(void* const* d_in, const int* in_sizes, int n_in,
                              void* d_out, int out_size, void* d_ws, size_t ws_size,
                              hipStream_t stream) {
    const float* x   = (const float*)d_in[0];
    const float* a1p = (const float*)d_in[1];
    const float* a2p = (const float*)d_in[2];
    const float* a3p = (const float*)d_in[3];
    const int* batch = (const int*)d_in[4];
    // d_in[5] = num_segments lives on device; derive B from out_size instead.
    const int K = in_sizes[1];
    const int N = in_sizes[4];
    const int B = (out_size - N * DF) / DF;

    float* b1   = (float*)d_ws;                       // N
    float* b2   = b1 + N;                             // B*DF
    int* counts = (int*)(b2 + (size_t)B * DF);        // B
    int* starts = counts + B;                         // B+1
    float* coef = (float*)(starts + B + 1);           // 5K+1

    float* z     = (float*)d_out;                     // B*DF
    float* trans = z + (size_t)B * DF;                // N*DF

    long zero_n = (long)N + (long)B * DF + B;         // b1 | b2 | counts
    zero_kernel<<<(int)((zero_n + 255) / 256), 256, 0, stream>>>(
        (unsigned int*)d_ws, zero_n);
    count_kernel<<<(N + 255) / 256, 256, 0, stream>>>(batch, counts, N);
    scan_kernel<<<1, 1024, 0, stream>>>(counts, starts, B);
    coef_kernel<<<(K + 63) / 64, 64, 0, stream>>>(a1p, a2p, a3p, coef, K);
    for (int k = 0; k < K; ++k)
        iter_kernel<<<B, 128, 0, stream>>>(x, counts, starts, b1, b2, coef, K, k);
    final_kernel<<<B, 128, 0, stream>>>(x, starts, b1, b2, coef, K, z, trans);
}